// RHF_35708358099215
// MI455X (gfx1250) — compile-verified
//
#include <hip/hip_runtime.h>
#include <math.h>

typedef __attribute__((ext_vector_type(2))) float v2f;
typedef __attribute__((ext_vector_type(8))) float v8f;

#define PI_F 3.14159265358979323846f
#define TWO_PI_2_5 34.98683665524973f   /* 2 * pi^2.5 */

__device__ __forceinline__ float boys0(float x) {
  float xs  = fmaxf(x, 1e-12f);
  float big = 0.5f * sqrtf(PI_F / xs) * erff(sqrtf(xs));
  return (x < 1e-10f) ? (1.0f - x * (1.0f / 3.0f)) : big;
}

__device__ __forceinline__ float expo4(int i) {
  int t = i & 3;
  return (t == 0) ? 0.5f : (t == 1) ? 0.4f : (t == 2) ? 0.3f : 0.2f;
}

__device__ __forceinline__ v8f wmma4(v2f a, v2f b, v8f c) {
  // D = A(16x4,f32) * B(4x16,f32) + C(16x16,f32)  -> v_wmma_f32_16x16x4_f32
  return __builtin_amdgcn_wmma_f32_16x16x4_f32(false, a, false, b, (short)0, c,
                                               false, false);
}

// out(64) = Mat(64x64, LDS row-major) * v(64): 4 row-blocks of 16, K chunks of 4.
// B fragment carries v in column 0 only. Column masking is done with an
// unconditional aligned 8B LDS load * precomputed mask (avoids EXEC-predicated
// ds_loads on the critical chain). 4 independent WMMA accumulator chains.
__device__ __forceinline__ void wmma_matvec64(const float* __restrict__ Mat,
                                              const float* __restrict__ v,
                                              float* __restrict__ out, int lane) {
  const int half = lane >> 4, l16 = lane & 15;
  const float colmask = (l16 == 0) ? 1.0f : 0.0f;
  v8f c0 = {}, c1 = {}, c2 = {}, c3 = {};
#pragma unroll
  for (int k0 = 0; k0 < 64; k0 += 4) {
    const int k = k0 + 2 * half;              // even -> 8B aligned
    v2f bb = *(const v2f*)(v + k);            // unconditional ds_load_b64
    v2f b;
    b.x = bb.x * colmask;
    b.y = bb.y * colmask;
    v2f a0, a1, a2, a3;
    a0.x = Mat[(l16) * 64 + k];      a0.y = Mat[(l16) * 64 + k + 1];
    a1.x = Mat[(16 + l16) * 64 + k]; a1.y = Mat[(16 + l16) * 64 + k + 1];
    a2.x = Mat[(32 + l16) * 64 + k]; a2.y = Mat[(32 + l16) * 64 + k + 1];
    a3.x = Mat[(48 + l16) * 64 + k]; a3.y = Mat[(48 + l16) * 64 + k + 1];
    c0 = wmma4(a0, b, c0);
    c1 = wmma4(a1, b, c1);
    c2 = wmma4(a2, b, c2);
    c3 = wmma4(a3, b, c3);
  }
  if (l16 == 0) {
#pragma unroll
    for (int r = 0; r < 8; ++r) {
      out[ 0 + r + 8 * half] = c0[r];
      out[16 + r + 8 * half] = c1[r];
      out[32 + r + 8 * half] = c2[r];
      out[48 + r + 8 * half] = c3[r];
    }
  }
}

// D(16x16) = A(16x16) * B(16x16), all LDS row-major, via 4 chained 16x16x4 WMMAs.
__device__ __forceinline__ void wmma_mm16(const float* __restrict__ A,
                                          const float* __restrict__ B,
                                          float* __restrict__ D, int lane) {
  const int half = lane >> 4, l16 = lane & 15;
  v8f c = {};
#pragma unroll
  for (int k0 = 0; k0 < 16; k0 += 4) {
    const int k = k0 + 2 * half;
    v2f a, b;
    a.x = A[l16 * 16 + k];   a.y = A[l16 * 16 + k + 1];
    b.x = B[k * 16 + l16];   b.y = B[(k + 1) * 16 + l16];
    c = wmma4(a, b, c);
  }
#pragma unroll
  for (int r = 0; r < 8; ++r) D[(r + 8 * half) * 16 + l16] = c[r];
}

// Cyclic Jacobi eigensolver for 8x8 symmetric M (LDS, destroyed).
// V gets eigenvectors (columns), w eigenvalues. Rotations parallel over 8 lanes;
// rotation params uniform (read from LDS by all lanes). Single-wave WG: barriers cheap.
__device__ void jacobi8(float* M, float* V, float* w, int lane) {
  for (int t = lane; t < 64; t += 32) V[t] = ((t >> 3) == (t & 7)) ? 1.0f : 0.0f;
  __syncthreads();
  for (int sweep = 0; sweep < 10; ++sweep) {
    for (int p = 0; p < 7; ++p) {
      for (int q = p + 1; q < 8; ++q) {
        float apq = M[p * 8 + q];                 // uniform across lanes
        if (fabsf(apq) > 1e-13f) {                // uniform branch
          float app = M[p * 8 + p], aqq = M[q * 8 + q];
          float tau = (aqq - app) / (2.0f * apq);
          float t = ((tau >= 0.0f) ? 1.0f : -1.0f) /
                    (fabsf(tau) + sqrtf(1.0f + tau * tau));
          float c = 1.0f / sqrtf(1.0f + t * t);
          float s = t * c;
          if (lane < 8) {                         // column rotation (M and V)
            int k = lane;
            float mkp = M[k * 8 + p], mkq = M[k * 8 + q];
            M[k * 8 + p] = c * mkp - s * mkq;
            M[k * 8 + q] = s * mkp + c * mkq;
            float vkp = V[k * 8 + p], vkq = V[k * 8 + q];
            V[k * 8 + p] = c * vkp - s * vkq;
            V[k * 8 + q] = s * vkp + c * vkq;
          }
          __syncthreads();
          if (lane < 8) {                         // row rotation
            int k = lane;
            float mpk = M[p * 8 + k], mqk = M[q * 8 + k];
            M[p * 8 + k] = c * mpk - s * mqk;
            M[q * 8 + k] = s * mpk + c * mqk;
          }
          __syncthreads();
        }
      }
    }
  }
  for (int t = lane; t < 8; t += 32) w[t] = M[t * 8 + t];
  __syncthreads();
}

__global__ __launch_bounds__(32) void hf8_scf_kernel(const float* __restrict__ geom,
                                                     float* __restrict__ out) {
  __shared__ __align__(16) float Gj[4096];   // G[pq][rs] for J
  __shared__ __align__(16) float Gk[4096];   // G[p r q s] permuted for K
  __shared__ float pairK[64], Pxs[64], Pys[64], Pzs[64], gss[64];
  __shared__ float Hs[64];
  __shared__ __align__(16) float At[256], Ft[256], T1[256], Fpt[256]; // 16x16 tiles
  __shared__ __align__(16) float dvec[64];
  __shared__ float Jv[64], Kv[64];
  __shared__ float Mw[64], Vw[64], ww[8];
  __shared__ float scal[1];

  const int lane = threadIdx.x;
  const float g0x = geom[0], g0y = geom[1], g0z = geom[2];
  const float g1x = geom[3], g1y = geom[4], g1z = geom[5];

  for (int t = lane; t < 256; t += 32) { At[t] = 0.f; Ft[t] = 0.f; T1[t] = 0.f; Fpt[t] = 0.f; }
  for (int t = lane; t < 64; t += 32) dvec[t] = 0.f;

  // ---- pair quantities: S (into Mw), H = T + V, product centers P, pair prefactors ----
  for (int ij = lane; ij < 64; ij += 32) {
    int i = ij >> 3, j = ij & 7;
    float a = expo4(i), b = expo4(j);
    float cxi = (i < 4) ? g0x : g1x, cyi = (i < 4) ? g0y : g1y, czi = (i < 4) ? g0z : g1z;
    float cxj = (j < 4) ? g0x : g1x, cyj = (j < 4) ? g0y : g1y, czj = (j < 4) ? g0z : g1z;
    float gv = a + b;
    float dx = cxi - cxj, dy = cyi - cyj, dz = czi - czj;
    float R2 = dx * dx + dy * dy + dz * dz;
    float ni = powf(2.0f * a / PI_F, 0.75f);
    float nj = powf(2.0f * b / PI_F, 0.75f);
    float abg = a * b / gv;
    float ee = expf(-abg * R2);
    float pk = ni * nj * ee;
    float pig = PI_F / gv;
    float S = pk * pig * sqrtf(pig);
    float T = abg * (3.0f - 2.0f * abg * R2) * S;
    float px = (a * cxi + b * cxj) / gv;
    float py = (a * cyi + b * cyj) / gv;
    float pz = (a * czi + b * czj) / gv;
    float pref = -(2.0f * PI_F / gv) * pk;
    float d0x = px - g0x, d0y = py - g0y, d0z = pz - g0z;
    float d1x = px - g1x, d1y = py - g1y, d1z = pz - g1z;
    float Vn = pref * boys0(gv * (d0x * d0x + d0y * d0y + d0z * d0z)) +
               pref * boys0(gv * (d1x * d1x + d1y * d1y + d1z * d1z));
    Hs[ij] = T + Vn;
    Mw[ij] = S;
    pairK[ij] = pk;
    Pxs[ij] = px; Pys[ij] = py; Pzs[ij] = pz; gss[ij] = gv;
  }
  __syncthreads();

  // ---- two-electron integrals G, 4096 elements (128 per lane) ----
  for (int t = lane; t < 4096; t += 32) {
    int pq = t >> 6, rs = t & 63;
    float g1v = gss[pq], g2v = gss[rs];
    float dx = Pxs[pq] - Pxs[rs];
    float dy = Pys[pq] - Pys[rs];
    float dz = Pzs[pq] - Pzs[rs];
    float PQ2 = dx * dx + dy * dy + dz * dz;
    float gs = g1v + g2v;
    float rho = g1v * g2v / gs;
    Gj[t] = pairK[pq] * pairK[rs] * TWO_PI_2_5 / (g1v * g2v * sqrtf(gs)) *
            boys0(rho * PQ2);
  }
  __syncthreads();
  for (int t = lane; t < 4096; t += 32) {   // K permutation: Gk[pq][rs] = G[p r q s]
    int pq = t >> 6, rs = t & 63;
    int p = pq >> 3, q = pq & 7, r = rs >> 3, s = rs & 7;
    Gk[t] = Gj[((p << 3) | r) * 64 + ((q << 3) | s)];
  }
  __syncthreads();

  // ---- A = S^(-1/2) via Jacobi (order-independent spectral function) ----
  jacobi8(Mw, Vw, ww, lane);
  for (int ij = lane; ij < 64; ij += 32) {
    int i = ij >> 3, j = ij & 7;
    float s = 0.0f;
#pragma unroll
    for (int k = 0; k < 8; ++k) s += Vw[i * 8 + k] * Vw[j * 8 + k] * rsqrtf(ww[k]);
    At[i * 16 + j] = s;   // top-left 8x8 of padded tile
  }
  __syncthreads();

  float ddx = g0x - g1x, ddy = g0y - g1y, ddz = g0z - g1z;
  float Enuc = rsqrtf(ddx * ddx + ddy * ddy + ddz * ddz);

  // ---- SCF loop ----
#pragma unroll 1
  for (int iter = 0; iter < 20; ++iter) {
    wmma_matvec64(Gj, dvec, Jv, lane);   // J = G  . D
    wmma_matvec64(Gk, dvec, Kv, lane);   // K = G' . D
    __syncthreads();
    for (int ij = lane; ij < 64; ij += 32)
      Ft[(ij >> 3) * 16 + (ij & 7)] = Hs[ij] + 2.0f * Jv[ij] - Kv[ij];
    __syncthreads();
    wmma_mm16(Ft, At, T1, lane);   // T1 = F * A
    __syncthreads();
    wmma_mm16(At, T1, Fpt, lane);  // Fp = A * T1
    __syncthreads();
    for (int ij = lane; ij < 64; ij += 32)
      Mw[ij] = Fpt[(ij >> 3) * 16 + (ij & 7)];
    __syncthreads();
    jacobi8(Mw, Vw, ww, lane);
    if (lane == 0) {
      int idx = 0; float best = ww[0];
#pragma unroll
      for (int k = 1; k < 8; ++k) if (ww[k] < best) { best = ww[k]; idx = k; }
      float cv[8];
#pragma unroll
      for (int i = 0; i < 8; ++i) {
        float s = 0.0f;
        for (int k = 0; k < 8; ++k) s += At[i * 16 + k] * Vw[k * 8 + idx];
        cv[i] = s;                          // C_occ = A @ v_min  (NDOCC == 1)
      }
      float E = 0.0f;
      for (int i = 0; i < 8; ++i)
        for (int j = 0; j < 8; ++j) {
          float dn = cv[i] * cv[j];        // Dn = Cocc Cocc^T
          dvec[i * 8 + j] = dn;
          E += (Ft[i * 16 + j] + Hs[i * 8 + j]) * dn;
        }
      scal[0] = E + Enuc;
    }
    __syncthreads();
  }
  if (lane == 0) out[0] = scal[0];
}

extern "C" void kernel_launch(void* const* d_in, const int* in_sizes, int n_in,
                              void* d_out, int out_size, void* d_ws, size_t ws_size,
                              hipStream_t stream) {
  (void)in_sizes; (void)n_in; (void)out_size; (void)d_ws; (void)ws_size;
  const float* geom = (const float*)d_in[0];   // 6 floats (2x3), f32
  float* out = (float*)d_out;                  // scalar f32 energy
  hipLaunchKernelGGL(hf8_scf_kernel, dim3(1), dim3(32), 0, stream, geom, out);
}